// CRF_10007273799899
// MI455X (gfx1250) — compile-verified
//
#include <hip/hip_runtime.h>
#include <stdint.h>

// ---------------- problem constants ----------------
#define B_DIM 4096
#define T_DIM 2048
#define K_TAGS 6
#define START_TAG 4
#define STOP_TAG 5
#define NEGV (-10000.0f)
#define NINF (-1.0e30f)

// ---------------- tiling ----------------
#define CHUNKS 32            // parallel-scan chunks along T
#define CLEN   64            // T / CHUNKS
#define TS     4             // time steps staged per async tile (96B per lane)
#define NTILES (CLEN / TS)   // 16
#define BLK    256
#define BPC    (B_DIM / BLK) // 16 blocks per chunk

// padded floats per lane per staging row: >= TS*6 = 24, multiple of 4 (16B LDS
// alignment for b128 async writes), stride 28 words -> reduced bank conflicts.
#define FSTRIDE 28

// ---------------- CDNA5 async global->LDS (gfx1250) ----------------
// GVS mode: mem = SGPR64 + VGPR32 + offset ; LDS dst = VDST-vgpr + offset.
#define ASYNC_B128(LDS, VOFF, SBASE, IMM)                                  \
  asm volatile("global_load_async_to_lds_b128 %0, %1, %2 offset:" #IMM     \
               :: "v"(LDS), "v"(VOFF), "s"(SBASE) : "memory")

__device__ __forceinline__ void issue_feat_tile(unsigned lds, unsigned voff,
                                                unsigned long long sbase) {
  // Async LDS writes are unordered vs this wave's in-flight DS reads; drain
  // DS before the engine may overwrite a staging buffer.
  asm volatile("s_wait_dscnt 0" ::: "memory");
  ASYNC_B128(lds, voff, sbase, 0);
  ASYNC_B128(lds, voff, sbase, 16);
  ASYNC_B128(lds, voff, sbase, 32);
  ASYNC_B128(lds, voff, sbase, 48);
  ASYNC_B128(lds, voff, sbase, 64);
  ASYNC_B128(lds, voff, sbase, 80);
}

// ---------------- pruned max-plus step helpers ----------------
// Alive exit/next rows: {0,1,2,3,STOP}. Row START stays NINF (can never win
// after t=0: margin ~1e4 from the NEG=-10000 structural constraints).
// Alive intermediate tags q: {0,1,2,3} (q=START row is NINF, q=STOP is
// forbidden by trans[:,STOP]=NEG, losing by >=1e4).

__device__ __forceinline__ void init_step(
    float (&Md)[36], const float (&f)[6], bool inter,
    const float (&tiv)[36], const float (&tav)[36])
{
  const int R[5] = {0, 1, 2, 3, 5};
#pragma unroll
  for (int ni = 0; ni < 5; ++ni) {
    const int n = R[ni];
#pragma unroll
    for (int p = 0; p < 6; ++p) {
      const float s = inter ? tiv[n * 6 + p] : tav[n * 6 + p];
      Md[n * 6 + p] = s + f[n];
    }
  }
}

__device__ __forceinline__ void compose_step(
    const float (&Ms)[36], float (&Md)[36], const float (&f)[6], bool inter,
    const float (&tiv)[36], const float (&tav)[36])
{
  const int R[5] = {0, 1, 2, 3, 5};
#pragma unroll
  for (int ni = 0; ni < 5; ++ni) {
    const int n = R[ni];
    float sr[4];
#pragma unroll
    for (int q = 0; q < 4; ++q)
      sr[q] = inter ? tiv[n * 6 + q] : tav[n * 6 + q];
#pragma unroll
    for (int p = 0; p < 6; ++p) {
      float m = sr[0] + Ms[0 * 6 + p];
#pragma unroll
      for (int q = 1; q < 4; ++q) m = fmaxf(m, sr[q] + Ms[q * 6 + p]);
      Md[n * 6 + p] = m + f[n];
    }
  }
}

// =====================================================================
// Kernel 1: per (b, chunk) compose the chunk's 64 Viterbi steps into a
// 6x6 max-plus affine matrix M (feats folded in). wsM layout [c][i][b].
// Register ping-pong: step parity == (s & 1) since TS is even, so no
// per-step 36-register copy is needed.
// =====================================================================
__global__ __launch_bounds__(BLK) void crf_k1_chunk_matrix(
    const float* __restrict__ feats, const int* __restrict__ chg,
    const float* __restrict__ ti, const float* __restrict__ ta,
    float* __restrict__ wsM)
{
  __shared__ __align__(16) float fst[2][BLK][FSTRIDE];
  const int tid = threadIdx.x;
  const int c   = blockIdx.x / BPC;
  const int b   = (blockIdx.x % BPC) * BLK + tid;

  float tiv[36], tav[36];           // uniform -> SGPRs
#pragma unroll
  for (int i = 0; i < 36; ++i) { tiv[i] = ti[i]; tav[i] = ta[i]; }

  float M[36], Mn[36];
#pragma unroll
  for (int i = 0; i < 36; ++i) { M[i] = NINF; Mn[i] = NINF; }

  const unsigned long long sb =
      (unsigned long long)(uintptr_t)feats +
      (unsigned long long)c * CLEN * K_TAGS * 4ull;           // uniform/block
  unsigned vo = (unsigned)b * (unsigned)(T_DIM * K_TAGS * 4); // per-lane
  const unsigned lds0 = (unsigned)(uintptr_t)&fst[0][tid][0];
  const unsigned lds1 = (unsigned)(uintptr_t)&fst[1][tid][0];
  const int* chgp = chg + (size_t)b * T_DIM + (size_t)c * CLEN;

  issue_feat_tile(lds0, vo, sb);
  vo += TS * K_TAGS * 4;

  for (int tile = 0; tile < NTILES; ++tile) {
    const int4 cg4 = *(const int4*)(chgp + tile * TS);
    if (tile + 1 < NTILES) {
      issue_feat_tile(((tile + 1) & 1) ? lds1 : lds0, vo, sb);
      vo += TS * K_TAGS * 4;
      asm volatile("s_wait_asynccnt 6" ::: "memory"); // prev tile landed
    } else {
      asm volatile("s_wait_asynccnt 0" ::: "memory");
    }
    const float* fb = &fst[tile & 1][tid][0];
    const int cgs[4] = {cg4.x, cg4.y, cg4.z, cg4.w};
#pragma unroll
    for (int s = 0; s < TS; ++s) {
      float f[6];
#pragma unroll
      for (int j = 0; j < 6; ++j) f[j] = fb[s * 6 + j];
      const bool inter = cgs[s] > 0;
      if ((s & 1) == 0) {                     // even global step: M -> Mn
        if (tile == 0 && s == 0) init_step(Mn, f, inter, tiv, tav);
        else                     compose_step(M, Mn, f, inter, tiv, tav);
      } else {                                // odd global step: Mn -> M
        compose_step(Mn, M, f, inter, tiv, tav);
      }
    }
  }

  // CLEN even -> final state lives in M (row START is NINF by construction)
  float* dst = wsM + (size_t)c * 36 * B_DIM + b;   // coalesced across lanes
#pragma unroll
  for (int i = 0; i < 36; ++i) dst[(size_t)i * B_DIM] = M[i];
}

// =====================================================================
// Kernel 2: per b, sequentially combine the 32 chunk matrices. Stores the
// entry alpha of every chunk, the path score, and the optimal tag at the
// end of every chunk (chunk-granular backtrack).
// =====================================================================
__global__ __launch_bounds__(BLK) void crf_k2_combine(
    const float* __restrict__ ta, const float* __restrict__ wsM,
    float* __restrict__ wsAlpha, int* __restrict__ wsEtag,
    float* __restrict__ out_score)
{
  __shared__ int cb[BLK][CHUNKS + 1];    // padded: conflict-free
  const int tid = threadIdx.x;
  const int b   = blockIdx.x * BLK + tid;

  float a[6];
#pragma unroll
  for (int n = 0; n < 6; ++n) a[n] = NEGV;
  a[START_TAG] = 0.0f;

  for (int c0 = 0; c0 < CHUNKS; ++c0) {
#pragma unroll
    for (int n = 0; n < 6; ++n)
      wsAlpha[((size_t)c0 * 6 + n) * B_DIM + b] = a[n];   // entry alpha
    float Mv[36];
#pragma unroll
    for (int i = 0; i < 36; ++i)
      Mv[i] = wsM[((size_t)c0 * 36 + i) * B_DIM + b];
    float an[6]; int w = 0;
#pragma unroll
    for (int n = 0; n < 6; ++n) {
      float best = Mv[n * 6 + 0] + a[0]; int bi = 0;
#pragma unroll
      for (int p = 1; p < 6; ++p) {
        float cnd = Mv[n * 6 + p] + a[p];
        if (cnd > best) { best = cnd; bi = p; }  // first-index ties, like argmax
      }
      an[n] = best; w |= bi << (3 * n);
    }
#pragma unroll
    for (int n = 0; n < 6; ++n) a[n] = an[n];
    cb[tid][c0] = w;
  }

  // terminal: STOP row of the intra matrix
  float best = a[0] + ta[STOP_TAG * 6 + 0]; int bt = 0;
#pragma unroll
  for (int n = 1; n < 6; ++n) {
    float cnd = a[n] + ta[STOP_TAG * 6 + n];
    if (cnd > best) { best = cnd; bt = n; }
  }
  out_score[b] = best;

  int e = bt;                                    // tag at end of last chunk
  for (int c0 = CHUNKS - 1; c0 >= 0; --c0) {
    wsEtag[(size_t)c0 * B_DIM + b] = e;
    e = (cb[tid][c0] >> (3 * e)) & 7;            // entry tag = end of c0-1
  }
}

// =====================================================================
// Kernel 3: per (b, chunk) re-run the chunk from its exact entry alpha,
// record packed 3-bit backpointers in LDS, then backtrack from the known
// chunk-end tag and emit the path (as float, per d_out convention).
// Prev candidates pruned to {0,1,2,3,START}: prev=STOP is structurally
// forbidden (trans[:,STOP]=NEG); START is kept so chunk-0 step-0 picks
// the init tag.
// =====================================================================
__global__ __launch_bounds__(BLK) void crf_k3_backtrack(
    const float* __restrict__ feats, const int* __restrict__ chg,
    const float* __restrict__ ti, const float* __restrict__ ta,
    const float* __restrict__ wsAlpha, const int* __restrict__ wsEtag,
    float* __restrict__ out_path)
{
  __shared__ __align__(16) float fst[2][BLK][FSTRIDE];
  __shared__ int bpl[BLK][CLEN + 1];             // padded: conflict-free
  const int tid = threadIdx.x;
  const int c   = blockIdx.x / BPC;
  const int b   = (blockIdx.x % BPC) * BLK + tid;

  float tiv[36], tav[36];
#pragma unroll
  for (int i = 0; i < 36; ++i) { tiv[i] = ti[i]; tav[i] = ta[i]; }

  float a[6];
#pragma unroll
  for (int n = 0; n < 6; ++n)
    a[n] = wsAlpha[((size_t)c * 6 + n) * B_DIM + b];

  const unsigned long long sb =
      (unsigned long long)(uintptr_t)feats +
      (unsigned long long)c * CLEN * K_TAGS * 4ull;
  unsigned vo = (unsigned)b * (unsigned)(T_DIM * K_TAGS * 4);
  const unsigned lds0 = (unsigned)(uintptr_t)&fst[0][tid][0];
  const unsigned lds1 = (unsigned)(uintptr_t)&fst[1][tid][0];
  const int* chgp = chg + (size_t)b * T_DIM + (size_t)c * CLEN;

  issue_feat_tile(lds0, vo, sb);
  vo += TS * K_TAGS * 4;

  for (int tile = 0; tile < NTILES; ++tile) {
    const int4 cg4 = *(const int4*)(chgp + tile * TS);
    if (tile + 1 < NTILES) {
      issue_feat_tile(((tile + 1) & 1) ? lds1 : lds0, vo, sb);
      vo += TS * K_TAGS * 4;
      asm volatile("s_wait_asynccnt 6" ::: "memory");
    } else {
      asm volatile("s_wait_asynccnt 0" ::: "memory");
    }
    const float* fb = &fst[tile & 1][tid][0];
    const int cgs[4] = {cg4.x, cg4.y, cg4.z, cg4.w};
#pragma unroll
    for (int s = 0; s < TS; ++s) {
      float f[6];
#pragma unroll
      for (int j = 0; j < 6; ++j) f[j] = fb[s * 6 + j];
      const bool inter = cgs[s] > 0;
      const int R[5] = {0, 1, 2, 3, 5};
      float an[6]; int w = 0;
#pragma unroll
      for (int ni = 0; ni < 5; ++ni) {
        const int n = R[ni];
        float sr0 = inter ? tiv[n * 6 + 0] : tav[n * 6 + 0];
        float best = sr0 + a[0]; int bi = 0;
#pragma unroll
        for (int p = 1; p < 5; ++p) {          // p=0..4 (START kept, STOP pruned)
          const float sr = inter ? tiv[n * 6 + p] : tav[n * 6 + p];
          const float cnd = sr + a[p];
          if (cnd > best) { best = cnd; bi = p; }
        }
        an[n] = best + f[n]; w |= bi << (3 * n);
      }
      an[START_TAG] = NINF;                    // START unreachable after t=0
#pragma unroll
      for (int n = 0; n < 6; ++n) a[n] = an[n];
      bpl[tid][tile * TS + s] = w;
    }
  }

  int tag = wsEtag[(size_t)c * B_DIM + b];       // tag at last step of chunk
  float* po = out_path + (size_t)b * T_DIM + (size_t)c * CLEN;
  for (int t = CLEN - 1; t >= 0; --t) {
    po[t] = (float)tag;
    tag = (bpl[tid][t] >> (3 * tag)) & 7;
  }
}

// =====================================================================
extern "C" void kernel_launch(void* const* d_in, const int* in_sizes, int n_in,
                              void* d_out, int out_size, void* d_ws, size_t ws_size,
                              hipStream_t stream) {
  (void)in_sizes; (void)n_in; (void)out_size; (void)ws_size;
  const float* feats = (const float*)d_in[0];
  const int*   chg   = (const int*)d_in[1];
  const float* ti    = (const float*)d_in[2];
  const float* ta    = (const float*)d_in[3];
  float* out = (float*)d_out;

  float* wsM     = (float*)d_ws;                          // [C][36][B]
  float* wsAlpha = wsM + (size_t)CHUNKS * 36 * B_DIM;     // [C][6][B]
  int*   wsEtag  = (int*)(wsAlpha + (size_t)CHUNKS * 6 * B_DIM); // [C][B]

  crf_k1_chunk_matrix<<<CHUNKS * BPC, BLK, 0, stream>>>(feats, chg, ti, ta, wsM);
  crf_k2_combine<<<B_DIM / BLK, BLK, 0, stream>>>(ta, wsM, wsAlpha, wsEtag, out);
  crf_k3_backtrack<<<CHUNKS * BPC, BLK, 0, stream>>>(feats, chg, ti, ta,
                                                     wsAlpha, wsEtag, out + B_DIM);
}